// BNNBranch_81106162417901
// MI455X (gfx1250) — compile-verified
//
#include <hip/hip_runtime.h>
#include <hip/hip_bf16.h>

typedef float v2f __attribute__((ext_vector_type(2)));
typedef float v8f __attribute__((ext_vector_type(8)));

#define Bsz   64
#define Lsz   16384
#define Csz   128
#define Ksz   7
#define Lc    16378          // L - K + 1
#define Lp    4094           // Lc / 4
#define NPOS  (Bsz * Lc)     // 1,048,192 BN population
#define NBLK_STATS 512
#define STRIPES 128          // ceil(Lp / 32) pooled-position stripes per batch

__device__ __forceinline__ float sgnf(float v) {
    return (v > 0.f) ? 1.f : ((v < 0.f) ? -1.f : 0.f);
}

// ---------------------------------------------------------------------------
// Kernel A: deterministic partial sums of S_k (7) and Q_{k<=k'} (28) over all
// (b, l) conv positions. One 36-float record per block in ws.
// ---------------------------------------------------------------------------
__global__ __launch_bounds__(256) void bnn_stats_kernel(
    const float* __restrict__ x, float* __restrict__ partial)
{
    float s[7]  = {0.f,0.f,0.f,0.f,0.f,0.f,0.f};
    float q[28] = {};

    const int tid    = blockIdx.x * blockDim.x + threadIdx.x;
    const int stride = gridDim.x * blockDim.x;

    for (int idx = tid; idx < NPOS; idx += stride) {
        const int b = idx / Lc;
        const int l = idx - b * Lc;
        const float* xp = x + (size_t)b * Lsz + l;
        float v[7];
#pragma unroll
        for (int k = 0; k < 7; ++k) v[k] = xp[k];
#pragma unroll
        for (int k = 0; k < 7; ++k) s[k] += v[k];
        int qi = 0;
#pragma unroll
        for (int k = 0; k < 7; ++k)
#pragma unroll
            for (int k2 = k; k2 < 7; ++k2) { q[qi] += v[k] * v[k2]; ++qi; }
    }

    // deterministic block tree-reduction of the 35 accumulators
    __shared__ float red[256];
    const int lt = threadIdx.x;
#pragma unroll 1
    for (int v = 0; v < 35; ++v) {
        red[lt] = (v < 7) ? s[v] : q[v - 7];
        __syncthreads();
        for (int off = 128; off > 0; off >>= 1) {
            if (lt < off) red[lt] += red[lt + off];
            __syncthreads();
        }
        if (lt == 0) partial[blockIdx.x * 36 + v] = red[0];
        __syncthreads();
    }
}

// ---------------------------------------------------------------------------
// Kernel B: single block. Reduce block partials (fixed order), then fold the
// binarized-weight quadratic form into per-channel affine a_c, b_c.
// ---------------------------------------------------------------------------
__global__ __launch_bounds__(128) void bnn_finalize_kernel(
    const float* __restrict__ partial, const float* __restrict__ w,
    const float* __restrict__ gamma,   const float* __restrict__ beta,
    float* __restrict__ ab)
{
    __shared__ float sums[35];
    const int t = threadIdx.x;
    if (t < 35) {
        float acc = 0.f;
        for (int i = 0; i < NBLK_STATS; ++i) acc += partial[i * 36 + t];
        sums[t] = acc;
    }
    __syncthreads();

    if (t < Csz) {
        float sg[7];
#pragma unroll
        for (int k = 0; k < 7; ++k) sg[k] = sgnf(w[t * 7 + k]);
        float Y1 = 0.f;
#pragma unroll
        for (int k = 0; k < 7; ++k) Y1 += sg[k] * sums[k];
        float Y2 = 0.f;
        int qi = 7;
#pragma unroll
        for (int k = 0; k < 7; ++k)
#pragma unroll
            for (int k2 = k; k2 < 7; ++k2) {
                float term = sg[k] * sg[k2] * sums[qi++];
                Y2 += (k2 == k) ? term : 2.f * term;
            }
        const float invN = 1.f / (float)NPOS;
        const float mean = Y1 * invN;
        const float var  = Y2 * invN - mean * mean;
        const float a    = gamma[t] * rsqrtf(var + 1e-5f);
        ab[t]        = a;
        ab[Csz + t]  = beta[t] - mean * a;
    }
}

// ---------------------------------------------------------------------------
// Kernel C: conv(sign(w)) as WMMA f32 16x16x4 GEMM + fused BN-affine + sign +
// maxpool4 + LDS transpose for lane-contiguous 128B stores.
// grid = B * STRIPES blocks of 256 (8 waves); wave = 16-channel tile.
// ---------------------------------------------------------------------------
__global__ __launch_bounds__(256) void bnn_conv_kernel(
    const float* __restrict__ x, const float* __restrict__ w,
    const float* __restrict__ ab, float* __restrict__ out)
{
    const int b  = blockIdx.x / STRIPES;
    const int s  = blockIdx.x - b * STRIPES;
    const int l0 = s * 128;   // raw conv-position base of this stripe
    const int p0 = s * 32;    // pooled-position base

    const int tid  = threadIdx.x;
    const int wave = tid >> 5;
    const int lane = tid & 31;
    const int half = lane >> 4;   // 0 | 1
    const int ln16 = lane & 15;

    __shared__ float xs[136];               // x window, zero-padded tail
    __shared__ float outs[8][16][33];       // [wave][c_local][p_local], padded

    for (int i = tid; i < 136; i += 256) {
        const int gi = l0 + i;
        xs[i] = (gi < Lsz) ? x[(size_t)b * Lsz + gi] : 0.f;
    }
    __syncthreads();

    // loop-invariant B operand: sign(weights), K padded 7 -> 8 with zero
    const int c0  = wave * 16;
    const int cch = c0 + ln16;
    v2f bm0, bm1;
    {
        const int k0 = 2 * half;            // K-half 0 covers K = 0..3
        bm0.x = sgnf(w[cch * 7 + k0]);
        bm0.y = sgnf(w[cch * 7 + k0 + 1]);
        const int k1 = 4 + 2 * half;        // K-half 1 covers K = 4..7
        bm1.x = sgnf(w[cch * 7 + k1]);
        bm1.y = (k1 + 1 < 7) ? sgnf(w[cch * 7 + k1 + 1]) : 0.f;
    }
    const float aff_a = ab[cch];
    const float aff_b = ab[Csz + cch];

#pragma unroll
    for (int m = 0; m < 8; ++m) {
        const int lrow = m * 16 + ln16;     // A-matrix row = position in stripe
        v2f a0, a1;
        a0.x = xs[lrow + 2 * half];
        a0.y = xs[lrow + 2 * half + 1];
        a1.x = xs[lrow + 4 + 2 * half];
        a1.y = xs[lrow + 4 + 2 * half + 1];

        v8f acc = {};
        acc = __builtin_amdgcn_wmma_f32_16x16x4_f32(
                  false, a0, false, bm0, (short)0, acc, false, false);
        acc = __builtin_amdgcn_wmma_f32_16x16x4_f32(
                  false, a1, false, bm1, (short)0, acc, false, false);

        // BN affine + sign; lane's channel is constant, rows = v + 8*half
        float sg[8];
#pragma unroll
        for (int v = 0; v < 8; ++v) sg[v] = sgnf(aff_a * acc[v] + aff_b);

        // maxpool4 is intra-lane: rows {0..3} and {4..7}
        const float pA = fmaxf(fmaxf(sg[0], sg[1]), fmaxf(sg[2], sg[3]));
        const float pB = fmaxf(fmaxf(sg[4], sg[5]), fmaxf(sg[6], sg[7]));
        const int pl = 4 * m + 2 * half;    // pooled index within stripe
        outs[wave][ln16][pl]     = pA;
        outs[wave][ln16][pl + 1] = pB;
    }
    __syncthreads();

    // coalesced stores: lanes span 32 consecutive pooled positions
    const size_t obase = (size_t)b * ((size_t)Csz * Lp);
    const int p = p0 + lane;
#pragma unroll
    for (int i = 0; i < 16; ++i) {
        if (p < Lp)
            out[obase + (size_t)(c0 + i) * Lp + p] = outs[wave][i][lane];
    }
}

// ---------------------------------------------------------------------------
extern "C" void kernel_launch(void* const* d_in, const int* in_sizes, int n_in,
                              void* d_out, int out_size, void* d_ws, size_t ws_size,
                              hipStream_t stream) {
    const float* x     = (const float*)d_in[0];
    const float* w     = (const float*)d_in[1];
    const float* gamma = (const float*)d_in[2];
    const float* beta  = (const float*)d_in[3];
    float* out = (float*)d_out;
    float* ws  = (float*)d_ws;

    float* partial = ws;                        // NBLK_STATS * 36 floats
    float* ab      = ws + NBLK_STATS * 36;      // a[128] ++ b[128]

    bnn_stats_kernel<<<NBLK_STATS, 256, 0, stream>>>(x, partial);
    bnn_finalize_kernel<<<1, 128, 0, stream>>>(partial, w, gamma, beta, ab);
    bnn_conv_kernel<<<Bsz * STRIPES, 256, 0, stream>>>(x, w, ab, out);
}